// DropoutBlock_8366596293053
// MI455X (gfx1250) — compile-verified
//
#include <hip/hip_runtime.h>

typedef float v2f __attribute__((ext_vector_type(2)));
typedef float v8f __attribute__((ext_vector_type(8)));

#define BN_EPS 1e-5f

// One wave = one 16-row M tile. Block = 256 threads = 8 waves = 128 rows.
// out = BN_ReLU( sum_{k=0..26} Gather(feat, nbr[:,k]) @ W[k] )
// Exact-fp32 V_WMMA_F32_16X16X4_F32 (16x16 C tile, K-step 4).
//
// fp32 WMMA operand layouts (wave32):
//   A (16x4):  lane L, vgpr j  -> element (M = L&15, K = j + 2*(L>>4))
//   B (4x16):  lane L, vgpr j  -> element (K = j + 2*(L>>4), N = L&15)
//   C (16x16): lane L, vgpr v  -> element (M = v + 8*(L>>4), N = L&15)
//
// Weights are staged into LDS pre-swizzled as sW[k][kk][pair][col][2] so each
// B fragment is one aligned ds_load_b64 straight into an adjacent VGPR pair
// (no cross-register marshalling movs):
//   sW[((k*(CIN/4)+kk)*2+p)*64 + c*2 + j] = W[k][4*kk+2*p+j][c]
template <int CIN>
__global__ __launch_bounds__(256) void sparse_conv_bn_relu(
    const float* __restrict__ feat,   // [N, CIN]
    const int*   __restrict__ nbr,    // [N, 27]
    const float* __restrict__ W,      // [27, CIN, 32]
    const float* __restrict__ gamma,  // [32]
    const float* __restrict__ beta,   // [32]
    const float* __restrict__ mean,   // [32]
    const float* __restrict__ var,    // [32]
    float*       __restrict__ out,    // [N, 32]
    int N)
{
    constexpr int K      = 27;
    constexpr int COUT   = 32;
    constexpr int WELEMS = K * CIN * COUT;

    extern __shared__ float sW[];     // swizzled [27][CIN/4][2][32][2]

    // Cooperative stage + swizzle of all 27 weight matrices into LDS.
    for (int i = threadIdx.x; i < WELEMS; i += 256) {
        const int c   = i & 31;                 // output column
        const int kin = (i >> 5) & (CIN - 1);   // input channel (CIN is pow2)
        const int k   = i / (32 * CIN);         // kernel offset
        const int kk  = kin >> 2;               // K-step (group of 4)
        const int p   = (kin >> 1) & 1;         // K-pair within step
        const int j   = kin & 1;                // element within pair
        sW[(((k * (CIN / 4) + kk) * 2 + p) * 32 + c) * 2 + j] = W[i];
    }
    __syncthreads();

    const int lane = threadIdx.x & 31;
    const int wave = threadIdx.x >> 5;
    const int hi   = lane >> 4;       // half-wave: K-pair select / M+8 select
    const int lo   = lane & 15;       // M row within tile / N col within tile

    const int m0 = (blockIdx.x * 8 + wave) * 16;
    if (m0 >= N) return;              // wave-uniform: EXEC stays all-ones

    v8f acc0 = {};                    // cols 0..15
    v8f acc1 = {};                    // cols 16..31

    int r = m0 + lo;
    if (r >= N) r = N - 1;            // clamp for ragged tail (stores guarded)
    const int* nrow = nbr + (long)r * K;

    for (int k = 0; k < K; ++k) {
        const int row = nrow[k];
        const float* frow = feat + (long)row * CIN;
        const float* wk   = sW + k * CIN * COUT;   // CIN*32 floats per offset

        #pragma unroll
        for (int kk = 0; kk < CIN / 4; ++kk) {
            // A fragment: one global_load_b64 gather (8B aligned)
            const v2f a = *(const v2f*)(frow + 4 * kk + 2 * hi);

            // B fragments: single aligned ds_load_b64 each (swizzled layout)
            const v2f* bp = (const v2f*)(wk + (kk * 2 + hi) * 64);
            const v2f b0 = bp[lo];
            const v2f b1 = bp[lo + 16];

            acc0 = __builtin_amdgcn_wmma_f32_16x16x4_f32(
                false, a, false, b0, (short)0, acc0, false, false);
            acc1 = __builtin_amdgcn_wmma_f32_16x16x4_f32(
                false, a, false, b1, (short)0, acc1, false, false);
        }
    }

    // Eval-mode BN + ReLU. Lane's channels are lo and 16+lo.
    const float s0 = gamma[lo]      * __frsqrt_rn(var[lo]      + BN_EPS);
    const float o0 = beta[lo]       - mean[lo]      * s0;
    const float s1 = gamma[16 + lo] * __frsqrt_rn(var[16 + lo] + BN_EPS);
    const float o1 = beta[16 + lo]  - mean[16 + lo] * s1;

    #pragma unroll
    for (int v = 0; v < 8; ++v) {
        const int m = m0 + v + 8 * hi;
        if (m < N) {
            out[(long)m * COUT + lo]      = fmaxf(acc0[v] * s0 + o0, 0.0f);
            out[(long)m * COUT + 16 + lo] = fmaxf(acc1[v] * s1 + o1, 0.0f);
        }
    }
}

extern "C" void kernel_launch(void* const* d_in, const int* in_sizes, int n_in,
                              void* d_out, int out_size, void* d_ws, size_t ws_size,
                              hipStream_t stream) {
    const float* x      = (const float*)d_in[0];
    const int*   nbr    = (const int*)  d_in[1];
    const float* W1     = (const float*)d_in[2];
    const float* gamma1 = (const float*)d_in[3];
    const float* beta1  = (const float*)d_in[4];
    const float* mean1  = (const float*)d_in[5];
    const float* var1   = (const float*)d_in[6];
    const float* W2     = (const float*)d_in[7];
    const float* gamma2 = (const float*)d_in[8];
    const float* beta2  = (const float*)d_in[9];
    const float* mean2  = (const float*)d_in[10];
    const float* var2   = (const float*)d_in[11];

    float* out = (float*)d_out;
    float* h   = (float*)d_ws;        // intermediate [N, 32] activations

    const int N      = in_sizes[0] / 16;     // x is [N, 16]
    const int tiles  = (N + 15) / 16;
    const int blocks = (tiles + 7) / 8;      // 8 waves (tiles) per block

    sparse_conv_bn_relu<16><<<blocks, 256, 27 * 16 * 32 * sizeof(float), stream>>>(
        x, nbr, W1, gamma1, beta1, mean1, var1, h, N);

    sparse_conv_bn_relu<32><<<blocks, 256, 27 * 32 * 32 * sizeof(float), stream>>>(
        h, nbr, W2, gamma2, beta2, mean2, var2, out, N);
}